// ScatterEncodedPathsToNodeEncodings_30657476559246
// MI455X (gfx1250) — compile-verified
//
#include <hip/hip_runtime.h>
#include <hip/hip_bf16.h>
#include <cstdint>

// ---------------------------------------------------------------------------
// MI455X (gfx1250) implementation.
//
// Bandwidth-bound problem: encoded_paths = 256MB streamed twice -> ~20us floor
// at 23.3 TB/s. GEMMs (~4 GFLOP total) are negligible -> use full-precision
// V_WMMA_F32_16X16X4_F32 (wave32, 16x16 f32 tile per wave, K step 4).
// q (10MB), update (10MB), Wq, gate_W all fit in 192MB L2, so gathered q rows
// and the f32 atomic-add scatter into update are L2-resident.
// ---------------------------------------------------------------------------

#define DD 256           // feature dim
typedef __attribute__((ext_vector_type(2))) float v2f;
typedef __attribute__((ext_vector_type(8))) float v8f;

__device__ __forceinline__ void atomicMaxFloat(float* addr, float val) {
    // standard signed/unsigned split trick (works across sign mix)
    if (val >= 0.0f) atomicMax((int*)addr, __float_as_int(val));
    else             atomicMin((unsigned int*)addr, __float_as_uint(val));
}

// ---------------- K0: init workspace (update=0, segmax=-inf, denom=0) -------
__global__ void init_kernel(float* __restrict__ update, float* __restrict__ segmax,
                            float* __restrict__ denom, int NND, int NN) {
    int stride = gridDim.x * blockDim.x;
    for (int j = blockIdx.x * blockDim.x + threadIdx.x; j < NND; j += stride)
        update[j] = 0.0f;
    for (int j = blockIdx.x * blockDim.x + threadIdx.x; j < NN; j += stride) {
        segmax[j] = -__builtin_inff();
        denom[j]  = 0.0f;
    }
}

// ---------------- K1: q = prev @ Wq  (WMMA f32 16x16x4) ---------------------
// grid = (D/16, ceil(mtiles/8)), block = 256 (8 waves); one 16x16 tile / wave.
__global__ __launch_bounds__(256)
void gemm_q_kernel(const float* __restrict__ prev, const float* __restrict__ Wq,
                   float* __restrict__ q, int NN) {
    __shared__ float Bs[DD * 16];                 // 16KB: Wq[:, n0:n0+16]
    const int n0  = blockIdx.x * 16;
    const int tid = threadIdx.x;
    for (int i = tid; i < DD * 16; i += 256) {
        int k = i >> 4, n = i & 15;
        Bs[i] = Wq[k * DD + n0 + n];
    }
    __syncthreads();

    const int wave  = tid >> 5;
    const int lane  = tid & 31;
    const int mtile = blockIdx.y * 8 + wave;
    if (mtile * 16 >= NN) return;
    const int m0    = mtile * 16;
    const int lrow  = lane & 15;
    const int khalf = (lane >> 4) * 2;            // 0 (lanes 0-15) or 2 (16-31)

    v8f acc = {};
    const float* aRow = prev + (size_t)(m0 + lrow) * DD;
    for (int k = 0; k < DD; k += 4) {
        v2f a; a.x = aRow[k + khalf]; a.y = aRow[k + khalf + 1];   // A 16x4 layout
        const float* bp = &Bs[(k + khalf) * 16 + lrow];
        v2f b; b.x = bp[0]; b.y = bp[16];                          // B 4x16 layout
        acc = __builtin_amdgcn_wmma_f32_16x16x4_f32(
            false, a, false, b, (short)0, acc, false, false);
    }
    // C/D layout: vgpr r -> row m0+r (lanes 0-15) / m0+8+r (lanes 16-31)
    const int col   = n0 + lrow;
    const int rbase = m0 + ((lane >> 4) << 3);
#pragma unroll
    for (int r = 0; r < 8; ++r)
        q[(size_t)(rbase + r) * DD + col] = acc[r];
}

// ---------------- K2: scores + atomic segment max ---------------------------
// one wave32 per path-step entry; lane handles 8 floats (2x float4)
__global__ __launch_bounds__(256)
void scores_kernel(const float* __restrict__ x, const uint8_t* __restrict__ mask,
                   const int* __restrict__ idx, const float* __restrict__ q,
                   float* __restrict__ scores, float* __restrict__ segmax, int N) {
    const int w    = (int)((blockIdx.x * blockDim.x + threadIdx.x) >> 5);
    const int lane = threadIdx.x & 31;
    if (w >= N) return;
    const int node = idx[w];
    const float4* xp = (const float4*)(x + (size_t)w * DD);
    const float4* qp = (const float4*)(q + (size_t)node * DD);
    float s = 0.0f;
#pragma unroll
    for (int step = 0; step < 2; ++step) {
        float4 xv = xp[lane + step * 32];
        float4 qv = qp[lane + step * 32];
        s += xv.x * qv.x + xv.y * qv.y + xv.z * qv.z + xv.w * qv.w;
    }
#pragma unroll
    for (int off = 16; off > 0; off >>= 1)
        s += __shfl_xor(s, off, 32);
    if (lane == 0) {
        s *= 0.0625f;                              // 1/sqrt(256)
        if (!mask[w]) s = -1e30f;
        scores[w] = s;
        atomicMaxFloat(&segmax[node], s);
    }
}

// ---------------- K3: segmax fixup (empty segments -> 0) --------------------
__global__ void fix_segmax_kernel(float* __restrict__ segmax, int NN) {
    int i = blockIdx.x * blockDim.x + threadIdx.x;
    if (i < NN) {
        float v = segmax[i];
        if (!(v > -__builtin_inff() && v < __builtin_inff())) segmax[i] = 0.0f;
    }
}

// ---------------- K4: e = exp(score - segmax)*m ; denom += e ----------------
__global__ void expsum_kernel(float* __restrict__ scores, const int* __restrict__ idx,
                              const uint8_t* __restrict__ mask,
                              const float* __restrict__ segmax,
                              float* __restrict__ denom, int N) {
    int i = blockIdx.x * blockDim.x + threadIdx.x;
    if (i >= N) return;
    const int node = idx[i];
    float ev = mask[i] ? __expf(scores[i] - segmax[node]) : 0.0f;
    scores[i] = ev;                                // reuse buffer as e[]
    if (ev != 0.0f) atomicAdd(&denom[node], ev);
}

// ---------------- K5: update[idx] += (e/denom) * x  (f32 global atomics) ----
__global__ __launch_bounds__(256)
void scatter_kernel(const float* __restrict__ x, const int* __restrict__ idx,
                    const float* __restrict__ e, const float* __restrict__ denom,
                    float* __restrict__ update, int N) {
    const int w    = (int)((blockIdx.x * blockDim.x + threadIdx.x) >> 5);
    const int lane = threadIdx.x & 31;
    if (w >= N) return;
    const float ev = e[w];
    if (ev == 0.0f) return;                        // masked: skip 2nd x read
    const int node = idx[w];
    const float p  = ev / fmaxf(denom[node], 1e-9f);
    const float4* xp = (const float4*)(x + (size_t)w * DD);
    float* up = update + (size_t)node * DD;
#pragma unroll
    for (int step = 0; step < 2; ++step) {
        float4 xv = xp[lane + step * 32];
        int base = (lane + step * 32) * 4;
        atomicAdd(up + base + 0, p * xv.x);
        atomicAdd(up + base + 1, p * xv.y);
        atomicAdd(up + base + 2, p * xv.z);
        atomicAdd(up + base + 3, p * xv.w);
    }
}

// ---------------- K6: gate GEMM (K=512, WMMA) + sigmoid + blend -------------
__global__ __launch_bounds__(256)
void gate_kernel(const float* __restrict__ prev, const float* __restrict__ update,
                 const float* __restrict__ gate_W, const float* __restrict__ gate_b,
                 float* __restrict__ out, int NN) {
    __shared__ float Bs[2 * DD * 16];              // 32KB: gate_W[:, n0:n0+16]
    const int n0  = blockIdx.x * 16;
    const int tid = threadIdx.x;
    for (int i = tid; i < 2 * DD * 16; i += 256) {
        int k = i >> 4, n = i & 15;
        Bs[i] = gate_W[k * DD + n0 + n];
    }
    __syncthreads();

    const int wave  = tid >> 5;
    const int lane  = tid & 31;
    const int mtile = blockIdx.y * 8 + wave;
    if (mtile * 16 >= NN) return;
    const int m0    = mtile * 16;
    const int lrow  = lane & 15;
    const int khalf = (lane >> 4) * 2;

    v8f acc = {};
    const float* aP = prev   + (size_t)(m0 + lrow) * DD;
    const float* aU = update + (size_t)(m0 + lrow) * DD;
    for (int k = 0; k < DD; k += 4) {              // first half of concat: prev
        v2f a; a.x = aP[k + khalf]; a.y = aP[k + khalf + 1];
        const float* bp = &Bs[(k + khalf) * 16 + lrow];
        v2f b; b.x = bp[0]; b.y = bp[16];
        acc = __builtin_amdgcn_wmma_f32_16x16x4_f32(
            false, a, false, b, (short)0, acc, false, false);
    }
    for (int k = 0; k < DD; k += 4) {              // second half: update
        v2f a; a.x = aU[k + khalf]; a.y = aU[k + khalf + 1];
        const float* bp = &Bs[(DD + k + khalf) * 16 + lrow];
        v2f b; b.x = bp[0]; b.y = bp[16];
        acc = __builtin_amdgcn_wmma_f32_16x16x4_f32(
            false, a, false, b, (short)0, acc, false, false);
    }

    const int col   = n0 + lrow;
    const int rbase = m0 + ((lane >> 4) << 3);
    const float bias = gate_b[col];
#pragma unroll
    for (int r = 0; r < 8; ++r) {
        const size_t o = (size_t)(rbase + r) * DD + col;
        float g  = 1.0f / (1.0f + __expf(-(acc[r] + bias)));
        out[o] = g * prev[o] + (1.0f - g) * update[o];
    }
}

// ---------------------------------------------------------------------------
extern "C" void kernel_launch(void* const* d_in, const int* in_sizes, int n_in,
                              void* d_out, int out_size, void* d_ws, size_t ws_size,
                              hipStream_t stream) {
    const float*   x    = (const float*)d_in[0];     // [4096,64,256]
    const uint8_t* mask = (const uint8_t*)d_in[1];   // [4096,64] bool (1B)
    const int*     idx  = (const int*)d_in[2];       // [4096,64]
    const float*   prev = (const float*)d_in[3];     // [NN,256]
    const float*   Wq   = (const float*)d_in[4];     // [256,256]
    const float*   gW   = (const float*)d_in[5];     // [512,256]
    const float*   gb   = (const float*)d_in[6];     // [256]
    float*         out  = (float*)d_out;

    const int N  = in_sizes[2];                      // 262144 path-steps
    const int NN = in_sizes[3] / DD;                 // 10000 nodes

    // workspace layout (floats): q | update | segmax | denom | scores
    float* ws     = (float*)d_ws;
    float* q      = ws;
    float* update = q + (size_t)NN * DD;
    float* segmax = update + (size_t)NN * DD;
    float* denom  = segmax + NN;
    float* scores = denom + NN;

    const int mtiles = NN / 16;                      // 10000/16 = 625
    dim3 gemmGrid(DD / 16, (mtiles + 7) / 8);        // (16, 79)

    init_kernel<<<1024, 256, 0, stream>>>(update, segmax, denom, NN * DD, NN);
    gemm_q_kernel<<<gemmGrid, 256, 0, stream>>>(prev, Wq, q, NN);
    scores_kernel<<<(N + 7) / 8, 256, 0, stream>>>(x, mask, idx, q, scores, segmax, N);
    fix_segmax_kernel<<<(NN + 255) / 256, 256, 0, stream>>>(segmax, NN);
    expsum_kernel<<<(N + 255) / 256, 256, 0, stream>>>(scores, idx, mask, segmax, denom, N);
    scatter_kernel<<<(N + 7) / 8, 256, 0, stream>>>(x, idx, scores, denom, update, N);
    gate_kernel<<<gemmGrid, 256, 0, stream>>>(prev, update, gW, gb, out, NN);
}